// LocalAttentionWindowModule_76948634075228
// MI455X (gfx1250) — compile-verified
//
#include <hip/hip_runtime.h>
#include <stdint.h>

typedef __attribute__((ext_vector_type(4))) unsigned int v4u;
typedef __attribute__((ext_vector_type(8))) int          v8i;
typedef __attribute__((ext_vector_type(4))) int          v4i;
typedef __attribute__((ext_vector_type(4))) float        v4f;

#ifndef __has_builtin
#define __has_builtin(x) 0
#endif
#if __has_builtin(__builtin_amdgcn_tensor_store_from_lds) && __has_builtin(__builtin_amdgcn_s_wait_tensorcnt)
#define HAVE_TDM 1
#else
#define HAVE_TDM 0
#endif

#define TILE_ELEMS 1024                       // 4 KB LDS zero tile (floats)
#define TDM_ITERS  64                         // iterations per TDM descriptor
#define FILL_ELEMS (TILE_ELEMS * TDM_ITERS)   // 65536 floats = 256 KB per WG = 4 rows of 16384

__global__ __launch_bounds__(256)
void attn_mask_tdm_kernel(const float* __restrict__ boxes,
                          float* __restrict__ out, int n)
{
    __shared__ float zbuf[TILE_ELEMS];
    const int t  = (int)threadIdx.x;
    const int wg = (int)blockIdx.x;
    const unsigned long long slab = (unsigned long long)wg * (unsigned long long)FILL_ELEMS;

    // ---- stage a 4 KB tile of zeros in LDS ----
    #pragma unroll
    for (int i = 0; i < TILE_ELEMS / 256; ++i)
        zbuf[t + 256 * i] = 0.0f;
    __syncthreads();

#if HAVE_TDM
    // ---- one wave issues a single TDM store: 64 iterations x 4 KB, LDS addr
    //      increment = 0 (reuse the zero tile), global increment = 4 KB ----
    if (t == 0) {
        const unsigned long long gaddr =
            (unsigned long long)(uintptr_t)out + slab * 4ull;
        const unsigned lds_off = (unsigned)(uintptr_t)&zbuf[0]; // low 32b of flat LDS addr = LDS offset

        v4u g0;
        g0.x = 1u;                                            // count=1 (valid user descriptor)
        g0.y = lds_off;                                       // lds_addr (bytes)
        g0.z = (unsigned)(gaddr & 0xffffffffull);             // global_addr[31:0]
        g0.w = (unsigned)((gaddr >> 32) & 0x01ffffffull)      // global_addr[56:32]
             | (2u << 30);                                    // type=2 ("image")

        v8i g1;
        g1[0] = (int)((2u << 16) | (1u << 19));               // data_size=4B, iterate_enable=1
        g1[1] = (int)((unsigned)TILE_ELEMS << 16);            // tensor_dim0[15:0]=1024
        g1[2] = (int)(1u << 16);                              // tensor_dim0 hi=0 | tensor_dim1[15:0]=1
        g1[3] = (int)((unsigned)TILE_ELEMS << 16);            // tensor_dim1 hi=0 | tile_dim0=1024
        g1[4] = 1;                                            // tile_dim1=1, tile_dim2=0
        g1[5] = TILE_ELEMS;                                   // tensor_dim0_stride = 1024
        g1[6] = 0;                                            // stride hi | dim1_stride lo = 0
        g1[7] = 0;

        v4i g2;
        g2[0] = 1;                                            // tensor_dim2 (unused with iterate)
        g2[1] = 0;                                            // lds_addr_increment = 0 (reuse zeros)
        g2[2] = TILE_ELEMS;                                   // global_addr_increment (data_size units)
        g2[3] = (int)((unsigned)(TDM_ITERS - 1) << 16);       // incr hi16=0 | iterate_count=63

        v4i g3 = {0, 0, 0, 0};                                // no dim3/dim4

#if __clang_major__ >= 23
        v8i g1b = {0, 0, 0, 0, 0, 0, 0, 0};
        __builtin_amdgcn_tensor_store_from_lds(g0, g1, g2, g3, g1b, 0);
#else
        __builtin_amdgcn_tensor_store_from_lds(g0, g1, g2, g3, 0);
#endif
        __builtin_amdgcn_s_wait_tensorcnt(0);
    }
    __syncthreads();   // all waves wait until TDM zero-fill of this slab has landed
#else
    // ---- fallback: vector NT zero-fill of this WG's 256 KB slab ----
    {
        v4f z = {0.0f, 0.0f, 0.0f, 0.0f};
        v4f* o4 = (v4f*)(out + slab);
        for (int i = t; i < FILL_ELEMS / 4; i += 256)
            __builtin_nontemporal_store(z, o4 + i);
    }
    __syncthreads();
#endif

    // ---- sparse overwrite: the band of ones (<= 99 per row) ----
    const int rows_per_wg = FILL_ELEMS / n;          // 4 for n = 16384
    if (rows_per_wg > 0) {
        const int tpr  = 256 / rows_per_wg;          // 64 threads per row
        const int r    = wg * rows_per_wg + t / tpr;
        const int lane = t % tpr;
        if (r < n) {
            const float w  = boxes[4 * r + 2];
            const float h  = boxes[4 * r + 3];
            const float mx = fmaxf(w, h);
            const float mn = fminf(w, h);
            const float scale = sqrtf(mx / mn);
            float wsf = floorf(33.0f * scale);
            wsf = fminf(fmaxf(wsf, 33.0f), 99.0f);
            const int half = ((int)wsf) >> 1;        // half in [16, 49]
            const int lo = r - half;
            int hi = r + half;
            if (hi > n - 1) hi = n - 1;
            const unsigned long long rb = (unsigned long long)r * (unsigned)n;
            for (int j = lo + lane; j <= hi; j += tpr)
                if (j >= 0)
                    __builtin_nontemporal_store(1.0f, out + rb + (unsigned)j);
        }
    }
}

extern "C" void kernel_launch(void* const* d_in, const int* in_sizes, int n_in,
                              void* d_out, int out_size, void* d_ws, size_t ws_size,
                              hipStream_t stream)
{
    (void)n_in; (void)d_ws; (void)ws_size; (void)out_size;
    const float* boxes = (const float*)d_in[0];
    float* out = (float*)d_out;
    const int n = in_sizes[0] / 4;                     // 16384
    const long long total = (long long)n * (long long)n;
    const int nwg = (int)((total + FILL_ELEMS - 1) / FILL_ELEMS);   // 4096
    attn_mask_tdm_kernel<<<nwg, 256, 0, stream>>>(boxes, out, n);
}